// _ProposalTargetLayer_44744969290563
// MI455X (gfx1250) — compile-verified
//
#include <hip/hip_runtime.h>
#include <hip/hip_bf16.h>

// ---------------- problem constants (match reference) ----------------
#define BIMG   32
#define NROI   12000
#define NGT    100
#define NTOT   12100          // NROI + NGT (gt appended as proposals)
#define RPI    512            // ROIS_PER_IMAGE
#define FGPI   128            // FG_PER_IMAGE
#define FG_THR 0.5f
#define NHALF  193600u        // (BIMG*NTOT)/2 for threefry counter split

// ---------------- JAX threefry2x32 (key = (0,42)) ----------------
__device__ __forceinline__ unsigned rotl32(unsigned x, int r) {
  return (x << r) | (x >> (32 - r));
}

// Reproduces jax.random.uniform(jax.random.key(42), (B, NTOT)) element `idx`
// (row-major flat index). JAX splits iota(n) into two halves as the two
// threefry counter words; out = concat(x0_half, x1_half).
__device__ float jax_uniform_at(unsigned idx) {
  const unsigned k0 = 0u, k1 = 42u;
  const unsigned ks2 = k0 ^ k1 ^ 0x1BD11BDAu;
  unsigned p  = (idx < NHALF) ? idx : idx - NHALF;
  unsigned x0 = p + k0;
  unsigned x1 = (p + NHALF) + k1;
#define TF_QR(r) { x0 += x1; x1 = rotl32(x1, r); x1 ^= x0; }
  TF_QR(13) TF_QR(15) TF_QR(26) TF_QR(6)  x0 += k1;  x1 += ks2 + 1u;
  TF_QR(17) TF_QR(29) TF_QR(16) TF_QR(24) x0 += ks2; x1 += k0  + 2u;
  TF_QR(13) TF_QR(15) TF_QR(26) TF_QR(6)  x0 += k0;  x1 += k1  + 3u;
  TF_QR(17) TF_QR(29) TF_QR(16) TF_QR(24) x0 += k1;  x1 += ks2 + 4u;
  TF_QR(13) TF_QR(15) TF_QR(26) TF_QR(6)  x0 += ks2; x1 += k0  + 5u;
#undef TF_QR
  unsigned bits = (idx < NHALF) ? x0 : x1;
  return __uint_as_float((bits >> 9) | 0x3f800000u) - 1.0f; // [0,1)
}

// ---------------- CDNA5 async global->LDS staging (B128) ----------------
// GLOBAL_LOAD_ASYNC_TO_LDS_B128 (gfx1250, tracked by ASYNCcnt). Generic
// pointer low 32 bits of a __shared__ address == wave-relative LDS byte
// address (flat-LDS aperture layout), which is what VDST expects.
__device__ __forceinline__ void async_g2l_b128(const void* gsrc, void* ldst) {
  unsigned lds_off = (unsigned)(unsigned long long)ldst;
  unsigned long long gaddr = (unsigned long long)gsrc;
  asm volatile("global_load_async_to_lds_b128 %0, %1, off"
               :: "v"(lds_off), "v"(gaddr) : "memory");
}
__device__ __forceinline__ void wait_async0() {
  asm volatile("s_wait_asynccnt 0" ::: "memory");
}

// ================= K1: IoU max/argmax + threefry u =================
// grid: (ceil(NTOT/256), BIMG), block 256
__global__ __launch_bounds__(256) void k_iou(const float* __restrict__ all_rois,
                                             const float* __restrict__ gt,
                                             float* __restrict__ maxov,
                                             int*   __restrict__ ga,
                                             float* __restrict__ uprio) {
  const int b   = blockIdx.y;
  const int tid = threadIdx.x;
  const int i   = blockIdx.x * 256 + tid;

  __shared__ __align__(16) float lgt[NGT * 5];
  // async-stage this image's GT table: 500 floats = 2000 B = 125 x B128,
  // one async transaction per lane of the first half-wavefronts
  if (tid < (NGT * 5) / 4)
    async_g2l_b128(gt + (size_t)b * NGT * 5 + tid * 4, &lgt[tid * 4]);
  wait_async0();
  __syncthreads();

  if (i >= NTOT) return;

  float x1, y1, x2, y2;
  if (i < NROI) {
    const float* p = all_rois + ((size_t)b * NROI + i) * 5;
    x1 = p[1]; y1 = p[2]; x2 = p[3]; y2 = p[4];
  } else {
    const int g = i - NROI;                 // appended gt proposal
    x1 = lgt[g * 5 + 0]; y1 = lgt[g * 5 + 1];
    x2 = lgt[g * 5 + 2]; y2 = lgt[g * 5 + 3];
  }
  const float area = (x2 - x1 + 1.0f) * (y2 - y1 + 1.0f);

  float best = -1.0f; int bi = 0;
  for (int g = 0; g < NGT; ++g) {
    const float gx1 = lgt[g * 5 + 0], gy1 = lgt[g * 5 + 1];
    const float gx2 = lgt[g * 5 + 2], gy2 = lgt[g * 5 + 3];
    const float ag = (gx2 - gx1 + 1.0f) * (gy2 - gy1 + 1.0f);
    float iw = fminf(x2, gx2) - fmaxf(x1, gx1) + 1.0f; iw = fmaxf(iw, 0.0f);
    float ih = fminf(y2, gy2) - fmaxf(y1, gy1) + 1.0f; ih = fmaxf(ih, 0.0f);
    const float inter = iw * ih;
    const float iou = inter / (area + ag - inter);
    if (iou > best) { best = iou; bi = g; }  // strict > == argmax first-max
  }
  const size_t e = (size_t)b * NTOT + i;
  maxov[e] = best;
  ga[e]    = bi;
  uprio[e] = jax_uniform_at((unsigned)e);
}

// ====== K2: per-image fg cap (128-th largest u among fg), prio ======
// grid: BIMG blocks, 256 threads
__global__ __launch_bounds__(256) void k_fgsel(const float* __restrict__ maxov,
                                               float* __restrict__ uprio,
                                               int*   __restrict__ fgcnt) {
  const int b = blockIdx.x, tid = threadIdx.x;
  const size_t base = (size_t)b * NTOT;
  __shared__ unsigned s_c;
  __shared__ int s_nfg;

  // count foreground
  if (tid == 0) s_nfg = 0;
  __syncthreads();
  int lc = 0;
  for (int i = tid; i < NTOT; i += 256) lc += (maxov[base + i] >= FG_THR);
  atomicAdd(&s_nfg, lc);
  __syncthreads();
  const int nfg = s_nfg;

  // binary search on float bits (u>=0 => bits order == value order) for the
  // largest T with count(fg && u_bits >= T) >= FGPI, i.e. the 128-th largest
  unsigned T = 0u;
  if (nfg > FGPI) {
    for (int bit = 30; bit >= 0; --bit) {
      const unsigned cand = T | (1u << bit);
      if (tid == 0) s_c = 0;
      __syncthreads();
      unsigned c = 0;
      for (int i = tid; i < NTOT; i += 256) {
        const size_t e = base + i;
        if (maxov[e] >= FG_THR && __float_as_uint(uprio[e]) >= cand) c++;
      }
      atomicAdd(&s_c, c);
      __syncthreads();
      if (s_c >= (unsigned)FGPI) T = cand;
      __syncthreads();
    }
  }
  if (tid == 0) fgcnt[b] = (nfg < FGPI) ? nfg : FGPI;

  // rewrite u -> priority in place: fg_sel: 2+u, bg(=!fg): 1+u, spilled fg: u
  for (int i = tid; i < NTOT; i += 256) {
    const size_t e = base + i;
    const float u = uprio[e];
    const bool fg = (maxov[e] >= FG_THR);
    const bool fsel = fg && (__float_as_uint(u) >= T);
    uprio[e] = fsel ? (2.0f + u) : (fg ? u : 1.0f + u);
  }
}

// ====== K3: per-image top-512 by prio, sorted desc (stable on idx) ======
// grid: BIMG blocks, 256 threads
__global__ __launch_bounds__(256) void k_topk(const float* __restrict__ prio,
                                              int* __restrict__ keep) {
  const int b = blockIdx.x, tid = threadIdx.x;
  const float* pr = prio + (size_t)b * NTOT;
  __shared__ unsigned s_c;
  __shared__ int s_n;
  __shared__ float sp[1024];
  __shared__ int   si[1024];

  // binary search the 512-th largest prio (prio in [0,3) => nonneg floats)
  unsigned T = 0u;
  for (int bit = 30; bit >= 0; --bit) {
    const unsigned cand = T | (1u << bit);
    if (tid == 0) s_c = 0;
    __syncthreads();
    unsigned c = 0;
    for (int i = tid; i < NTOT; i += 256)
      if (__float_as_uint(pr[i]) >= cand) c++;
    atomicAdd(&s_c, c);
    __syncthreads();
    if (s_c >= (unsigned)RPI) T = cand;
    __syncthreads();
  }

  // compact candidates (>=512, ties give a few extra; cap 1024)
  if (tid == 0) s_n = 0;
  __syncthreads();
  for (int i = tid; i < NTOT; i += 256) {
    const float p = pr[i];
    if (__float_as_uint(p) >= T) {
      const int slot = atomicAdd(&s_n, 1);
      if (slot < 1024) { sp[slot] = p; si[slot] = i; }
    }
  }
  __syncthreads();
  int n = s_n; if (n > 1024) n = 1024;
  for (int i = tid; i < 1024; i += 256)
    if (i >= n) { sp[i] = -1.0f; si[i] = 0x7FFFFFFF; }
  __syncthreads();

  // bitonic sort 1024: (prio desc, idx asc) == lax.top_k ordering
  for (int k = 2; k <= 1024; k <<= 1) {
    for (int j = k >> 1; j > 0; j >>= 1) {
      for (int t = tid; t < 1024; t += 256) {
        const int x = t ^ j;
        if (x > t) {
          const float pa = sp[t], pb = sp[x];
          const int ia = si[t], ib = si[x];
          const bool before = (pa > pb) || (pa == pb && ia < ib);
          const bool dirDesc = ((t & k) == 0);
          if (dirDesc != before) {
            sp[t] = pb; sp[x] = pa; si[t] = ib; si[x] = ia;
          }
        }
      }
      __syncthreads();
    }
  }
  for (int r = tid; r < RPI; r += 256) keep[(size_t)b * RPI + r] = si[r];
}

// ================= K4: gather + bbox transform + outputs =================
// grid: BIMG*RPI/256 blocks, 256 threads
__global__ __launch_bounds__(256) void k_out(const float* __restrict__ all_rois,
                                             const float* __restrict__ gt,
                                             const int* __restrict__ ga,
                                             const int* __restrict__ fgcnt,
                                             const int* __restrict__ keep,
                                             float* __restrict__ out) {
  const int t = blockIdx.x * 256 + threadIdx.x;  // 0 .. B*RPI-1
  const int b = t / RPI, r = t % RPI;
  const int i = keep[(size_t)b * RPI + r];

  float x1, y1, x2, y2;
  if (i < NROI) {
    const float* p = all_rois + ((size_t)b * NROI + i) * 5;
    x1 = p[1]; y1 = p[2]; x2 = p[3]; y2 = p[4];
  } else {
    const float* p = gt + ((size_t)b * NGT + (i - NROI)) * 5;
    x1 = p[0]; y1 = p[1]; x2 = p[2]; y2 = p[3];
  }
  const int g = ga[(size_t)b * NTOT + i];
  const float* q = gt + ((size_t)b * NGT + g) * 5;
  const float lbl = (r < fgcnt[b]) ? q[4] : 0.0f;

  const float ew = x2 - x1 + 1.0f, eh = y2 - y1 + 1.0f;
  const float ecx = x1 + 0.5f * ew, ecy = y1 + 0.5f * eh;
  const float gw = q[2] - q[0] + 1.0f, gh = q[3] - q[1] + 1.0f;
  const float gcx = q[0] + 0.5f * gw, gcy = q[1] + 0.5f * gh;
  float t0 = ((gcx - ecx) / ew) / 0.1f;
  float t1 = ((gcy - ecy) / eh) / 0.1f;
  float t2 = logf(gw / ew) / 0.2f;
  float t3 = logf(gh / eh) / 0.2f;

  const bool pos = (lbl > 0.0f);
  const float w = pos ? 1.0f : 0.0f;

  // output layout (flat float32, return order)
  float* o_rois = out;                                  // [B,512,5]
  float* o_lbl  = out + (size_t)BIMG * RPI * 5;         // [B,512]
  float* o_bt   = o_lbl + (size_t)BIMG * RPI;           // [B,512,4]
  float* o_biw  = o_bt  + (size_t)BIMG * RPI * 4;       // [B,512,4]
  float* o_bow  = o_biw + (size_t)BIMG * RPI * 4;       // [B,512,4]

  const size_t rr = (size_t)b * RPI + r;
  o_rois[rr * 5 + 0] = (float)b;
  o_rois[rr * 5 + 1] = x1; o_rois[rr * 5 + 2] = y1;
  o_rois[rr * 5 + 3] = x2; o_rois[rr * 5 + 4] = y2;
  o_lbl[rr] = lbl;
  o_bt[rr * 4 + 0] = pos ? t0 : 0.0f;
  o_bt[rr * 4 + 1] = pos ? t1 : 0.0f;
  o_bt[rr * 4 + 2] = pos ? t2 : 0.0f;
  o_bt[rr * 4 + 3] = pos ? t3 : 0.0f;
  o_biw[rr * 4 + 0] = w; o_biw[rr * 4 + 1] = w;
  o_biw[rr * 4 + 2] = w; o_biw[rr * 4 + 3] = w;
  o_bow[rr * 4 + 0] = w; o_bow[rr * 4 + 1] = w;
  o_bow[rr * 4 + 2] = w; o_bow[rr * 4 + 3] = w;
}

// ============================ launcher ============================
extern "C" void kernel_launch(void* const* d_in, const int* in_sizes, int n_in,
                              void* d_out, int out_size, void* d_ws, size_t ws_size,
                              hipStream_t stream) {
  (void)in_sizes; (void)n_in; (void)out_size; (void)ws_size;
  const float* all_rois = (const float*)d_in[0];   // [B,12000,5] f32
  const float* gt       = (const float*)d_in[1];   // [B,100,5]   f32
  // d_in[2] num_boxes unused (always == G in reference setup)
  float* out = (float*)d_out;

  // workspace layout (floats): maxov | ga | uprio | fgcnt | keep  (~4.7 MB)
  const size_t BN = (size_t)BIMG * NTOT;
  float* ws     = (float*)d_ws;
  float* maxov  = ws;
  int*   ga     = (int*)(ws + BN);
  float* uprio  = ws + 2 * BN;
  int*   fgcnt  = (int*)(ws + 3 * BN);
  int*   keep   = fgcnt + BIMG;

  dim3 g1((NTOT + 255) / 256, BIMG);
  k_iou  <<<g1, 256, 0, stream>>>(all_rois, gt, maxov, ga, uprio);
  k_fgsel<<<BIMG, 256, 0, stream>>>(maxov, uprio, fgcnt);
  k_topk <<<BIMG, 256, 0, stream>>>(uprio, keep);
  k_out  <<<(BIMG * RPI) / 256, 256, 0, stream>>>(all_rois, gt, ga, fgcnt, keep, out);
}